// Model_24524263260958
// MI455X (gfx1250) — compile-verified
//
#include <hip/hip_runtime.h>

typedef __attribute__((ext_vector_type(2))) float v2f;
typedef __attribute__((ext_vector_type(8))) float v8f;

#define NNODES 81
#define DIM 20
#define HID 100
#define GATE 400
#define BATCH 64
#define SEQ 2048
#define NEDGE 240
#define HPAD 101

__device__ __forceinline__ float fast_sigmoid(float x) {
  return 1.0f / (1.0f + __expf(-x));
}
__device__ __forceinline__ float fast_tanh(float x) {
  x = fminf(fmaxf(x, -15.0f), 15.0f);
  float e = __expf(2.0f * x);
  return (e - 1.0f) / (e + 1.0f);
}

// ---------------- GCN on the fixed 81-node graph (tiny; one block) ----------
__global__ void gcn_kernel(const float* __restrict__ emb_w,
                           const int* __restrict__ edge_index,
                           const float* __restrict__ W1, const float* __restrict__ b1,
                           const float* __restrict__ W2, const float* __restrict__ b2,
                           float* __restrict__ net_out) {
  __shared__ float deg[NNODES];
  __shared__ float dinv[NNODES];
  __shared__ float bufA[NNODES * DIM];
  __shared__ float bufB[NNODES * DIM];
  const int tid = threadIdx.x, bs = blockDim.x;
  const int* src = edge_index;
  const int* dst = edge_index + NEDGE;

  for (int i = tid; i < NNODES; i += bs) deg[i] = 1.0f;  // self loops
  __syncthreads();
  for (int e = tid; e < NEDGE; e += bs) atomicAdd(&deg[dst[e]], 1.0f);
  __syncthreads();
  for (int i = tid; i < NNODES; i += bs) dinv[i] = rsqrtf(deg[i]);
  __syncthreads();

  for (int idx = tid; idx < NNODES * DIM; idx += bs) {  // h1 = x @ W1
    int i = idx / DIM, j = idx % DIM;
    float s = 0.f;
    for (int k = 0; k < DIM; ++k) s += emb_w[i * DIM + k] * W1[k * DIM + j];
    bufA[idx] = s;
  }
  for (int idx = tid; idx < NNODES * DIM; idx += bs) bufB[idx] = 0.f;
  __syncthreads();
  for (int e = tid; e < NEDGE; e += bs) {  // aggregate h1
    int s = src[e], d = dst[e];
    float nm = dinv[s] * dinv[d];
    for (int j = 0; j < DIM; ++j) atomicAdd(&bufB[d * DIM + j], bufA[s * DIM + j] * nm);
  }
  for (int i = tid; i < NNODES; i += bs) {
    float nm = dinv[i] * dinv[i];
    for (int j = 0; j < DIM; ++j) atomicAdd(&bufB[i * DIM + j], bufA[i * DIM + j] * nm);
  }
  __syncthreads();
  for (int idx = tid; idx < NNODES * DIM; idx += bs)  // x2 = relu(agg + b1)
    bufA[idx] = fmaxf(bufB[idx] + b1[idx % DIM], 0.f);
  __syncthreads();
  for (int idx = tid; idx < NNODES * DIM; idx += bs) {  // h2 = x2 @ W2
    int i = idx / DIM, j = idx % DIM;
    float s = 0.f;
    for (int k = 0; k < DIM; ++k) s += bufA[i * DIM + k] * W2[k * DIM + j];
    bufB[idx] = s;
  }
  __syncthreads();
  for (int idx = tid; idx < NNODES * DIM; idx += bs) bufA[idx] = 0.f;
  __syncthreads();
  for (int e = tid; e < NEDGE; e += bs) {  // aggregate h2
    int s = src[e], d = dst[e];
    float nm = dinv[s] * dinv[d];
    for (int j = 0; j < DIM; ++j) atomicAdd(&bufA[d * DIM + j], bufB[s * DIM + j] * nm);
  }
  for (int i = tid; i < NNODES; i += bs) {
    float nm = dinv[i] * dinv[i];
    for (int j = 0; j < DIM; ++j) atomicAdd(&bufA[i * DIM + j], bufB[i * DIM + j] * nm);
  }
  __syncthreads();
  for (int idx = tid; idx < NNODES * DIM; idx += bs)
    net_out[idx] = bufA[idx] + b2[idx % DIM];
}

// ------- X0[b,t,0:20] = feats@fc_w^T + fc_b ; X0[b,t,20:40] = net[ids] ------
__global__ void build_x0(const float* __restrict__ feats, const int* __restrict__ ids,
                         const float* __restrict__ fc_w, const float* __restrict__ fc_b,
                         const float* __restrict__ net, float* __restrict__ X0, int total) {
  int idx = blockIdx.x * blockDim.x + threadIdx.x;
  if (idx >= total) return;
  float f0 = feats[(size_t)idx * 2], f1 = feats[(size_t)idx * 2 + 1];
  int id = ids[idx];
  float* o = X0 + (size_t)idx * 40;
  for (int j = 0; j < DIM; ++j) {
    o[j] = f0 * fc_w[j * 2] + f1 * fc_w[j * 2 + 1] + fc_b[j];
    o[DIM + j] = net[id * DIM + j];
  }
}

// --------- transpose in[R][C] -> out[C][R] (Wih prep, one-time) -------------
__global__ void transpose_kernel(const float* __restrict__ in, float* __restrict__ out,
                                 int R, int C) {
  int idx = blockIdx.x * blockDim.x + threadIdx.x;
  if (idx >= R * C) return;
  int r = idx / C, c = idx % C;
  out[c * R + r] = in[idx];
}

// ---------------------------------------------------------------------------
// BiLSTM scan. blockIdx.x: bit0 = direction (0=fwd,1=bwd), bits>1 = 16-row
// batch strip. 5 waves (160 threads); wave w owns the 5 contiguous N-tiles
// covering gate columns [80w, 80w+80) -> 25 tiles exactly, no guards, no EXEC
// divergence around WMMA. Whh^T lives in (dynamic) LDS; Wih^T streams from L2.
// fp32 WMMA 16x16x4 lane layout:
//   A (16x4): lanes 0-15 -> M=lane, K={k,k+1}; lanes 16-31 -> M=lane-16, K={k+2,k+3}
//   B (4x16): lanes 0-15 -> N=n0+lane, K={k,k+1}; lanes 16-31 -> K={k+2,k+3}
// ---------------------------------------------------------------------------
#define SCAN_THREADS 160
#define SCAN_SMEM_FLOATS (HID * GATE + 16 * GATE + 16 * HPAD + 16 * HID + GATE)

template <int KX>
__global__ void lstm_scan(const float* __restrict__ X,
                          const float* __restrict__ WihT_f, const float* __restrict__ Whh_f,
                          const float* __restrict__ bih_f, const float* __restrict__ bhh_f,
                          const float* __restrict__ WihT_b, const float* __restrict__ Whh_b,
                          const float* __restrict__ bih_b, const float* __restrict__ bhh_b,
                          float* __restrict__ out) {
  extern __shared__ float smem[];
  float* WHH = smem;                  // [HID][GATE] = Whh^T
  float* GATES = WHH + HID * GATE;    // [16][GATE]
  float* HS = GATES + 16 * GATE;      // [16][HPAD]
  float* CS = HS + 16 * HPAD;         // [16][HID]
  float* BIAS = CS + 16 * HID;        // [GATE]

  const int tid = threadIdx.x;
  const int lane = tid & 31;
  const int wave = tid >> 5;
  const int dir = blockIdx.x & 1;
  const int m0 = (blockIdx.x >> 1) * 16;
  const float* WihT = dir ? WihT_b : WihT_f;
  const float* Whh = dir ? Whh_b : Whh_f;
  const float* bih = dir ? bih_b : bih_f;
  const float* bhh = dir ? bhh_b : bhh_f;

  // one-time: transpose Whh into LDS, combine biases, zero state
  for (int i = tid; i < HID * GATE; i += SCAN_THREADS) {
    int k = i / GATE, n = i % GATE;
    WHH[i] = Whh[n * HID + k];
  }
  for (int i = tid; i < GATE; i += SCAN_THREADS) BIAS[i] = bih[i] + bhh[i];
  for (int i = tid; i < 16 * HID; i += SCAN_THREADS) CS[i] = 0.f;
  for (int i = tid; i < 16 * HPAD; i += SCAN_THREADS) HS[i] = 0.f;
  __syncthreads();

  const int l15 = lane & 15;
  const int ksub = (lane >> 4) * 2;
  const int hi8 = (lane >> 4) * 8;
  const int nbase = wave * 80 + l15;  // wave's first gate column for this lane
  const float* Xrow0 = X + (size_t)(m0 + l15) * (size_t)SEQ * (size_t)KX;

  for (int s = 0; s < SEQ; ++s) {
    const int t = dir ? (SEQ - 1 - s) : s;
    const float* xr = Xrow0 + (size_t)t * KX;
    v8f acc[5] = {};

    // ---- input projection: x_t @ Wih^T (B frags stream from L2) ----
#pragma unroll 5
    for (int k0 = 0; k0 < KX; k0 += 4) {
      v2f a;
      a.x = xr[k0 + ksub];
      a.y = xr[k0 + ksub + 1];
      const float* wr0 = WihT + (size_t)(k0 + ksub) * GATE + nbase;
      const float* wr1 = wr0 + GATE;
#pragma unroll
      for (int j = 0; j < 5; ++j) {
        v2f b;
        b.x = wr0[j * 16];
        b.y = wr1[j * 16];
        acc[j] = __builtin_amdgcn_wmma_f32_16x16x4_f32(false, a, false, b,
                                                       (short)0, acc[j], false, false);
      }
    }
    // ---- recurrence: h_{t-1} @ Whh^T (A and B frags from LDS) ----
#pragma unroll 5
    for (int k0 = 0; k0 < HID; k0 += 4) {
      v2f a;
      a.x = HS[l15 * HPAD + k0 + ksub];
      a.y = HS[l15 * HPAD + k0 + ksub + 1];
      const float* hr0 = WHH + (k0 + ksub) * GATE + nbase;
      const float* hr1 = hr0 + GATE;
#pragma unroll
      for (int j = 0; j < 5; ++j) {
        v2f b;
        b.x = hr0[j * 16];
        b.y = hr1[j * 16];
        acc[j] = __builtin_amdgcn_wmma_f32_16x16x4_f32(false, a, false, b,
                                                       (short)0, acc[j], false, false);
      }
    }
    // ---- gates (+bias) to LDS; C/D layout: VGPR r -> row r + hi8 ----
#pragma unroll
    for (int j = 0; j < 5; ++j) {
      int ncol = nbase + j * 16;
      float bv = BIAS[ncol];
#pragma unroll
      for (int r = 0; r < 8; ++r)
        GATES[(r + hi8) * GATE + ncol] = acc[j][r] + bv;
    }
    __syncthreads();
    // ---- LSTM cell update (i,f,g,o) ----
    for (int e = tid; e < 16 * HID; e += SCAN_THREADS) {
      int row = e / HID, n = e % HID;
      float gi = GATES[row * GATE + n];
      float gf = GATES[row * GATE + HID + n];
      float gg = GATES[row * GATE + 2 * HID + n];
      float go = GATES[row * GATE + 3 * HID + n];
      float iv = fast_sigmoid(gi), fv = fast_sigmoid(gf);
      float gv = fast_tanh(gg), ov = fast_sigmoid(go);
      float c = fv * CS[e] + iv * gv;
      float h = ov * fast_tanh(c);
      CS[e] = c;
      HS[row * HPAD + n] = h;
      out[((size_t)(m0 + row) * SEQ + t) * 200 + dir * HID + n] = h;
    }
    __syncthreads();
  }
}

// ---------------- MLP: relu(relu(H@W1^T+b1)@W2^T+b2) -> [B,T,2] -------------
__global__ void mlp_kernel(const float* __restrict__ H, const float* __restrict__ W1,
                           const float* __restrict__ B1, const float* __restrict__ W2,
                           const float* __restrict__ B2, float* __restrict__ out) {
  __shared__ float HIDS[16 * 56];
  const int tid = threadIdx.x, lane = tid & 31, wave = tid >> 5;
  const int m0 = blockIdx.x * 16;
  const int l15 = lane & 15;
  const int ksub = (lane >> 4) * 2;
  const int hi8 = (lane >> 4) * 8;
  const int ncol = wave * 16 + l15;
  const bool valid = (ncol < 50);
  const int nc = valid ? ncol : 0;  // clamp to avoid OOB loads
  const float msk = valid ? 1.f : 0.f;
  const float* arow = H + (size_t)(m0 + l15) * 200;
  v8f acc = {};
#pragma unroll 5
  for (int k0 = 0; k0 < 200; k0 += 4) {
    v2f a;
    a.x = arow[k0 + ksub];
    a.y = arow[k0 + ksub + 1];
    v2f b;
    b.x = W1[nc * 200 + k0 + ksub] * msk;
    b.y = W1[nc * 200 + k0 + ksub + 1] * msk;
    acc = __builtin_amdgcn_wmma_f32_16x16x4_f32(false, a, false, b,
                                                (short)0, acc, false, false);
  }
  if (valid) {
    float bv = B1[ncol];
#pragma unroll
    for (int r = 0; r < 8; ++r)
      HIDS[(r + hi8) * 56 + ncol] = fmaxf(acc[r] + bv, 0.f);
  }
  __syncthreads();
  if (tid < 32) {
    int row = tid >> 1, col = tid & 1;
    float s = B2[col];
    for (int j = 0; j < 50; ++j) s += HIDS[row * 56 + j] * W2[col * 50 + j];
    out[(size_t)(m0 + row) * 2 + col] = fmaxf(s, 0.f);
  }
}

extern "C" void kernel_launch(void* const* d_in, const int* in_sizes, int n_in,
                              void* d_out, int out_size, void* d_ws, size_t ws_size,
                              hipStream_t stream) {
  (void)in_sizes; (void)n_in; (void)out_size; (void)ws_size;
  const float* feats = (const float*)d_in[0];
  const int* ids = (const int*)d_in[1];
  // d_in[2] node_ids == arange, unused
  const int* edge_index = (const int*)d_in[3];
  const float* emb_w = (const float*)d_in[4];
  const float* gcn_w1 = (const float*)d_in[5];
  const float* gcn_b1 = (const float*)d_in[6];
  const float* gcn_w2 = (const float*)d_in[7];
  const float* gcn_b2 = (const float*)d_in[8];
  const float* fc_w = (const float*)d_in[9];
  const float* fc_b = (const float*)d_in[10];
  const float* Wih[4]; const float* Whh[4]; const float* bih[4]; const float* bhh[4];
  for (int p = 0; p < 4; ++p) {
    Wih[p] = (const float*)d_in[11 + 4 * p];
    Whh[p] = (const float*)d_in[12 + 4 * p];
    bih[p] = (const float*)d_in[13 + 4 * p];
    bhh[p] = (const float*)d_in[14 + 4 * p];
  }
  const float* mlp_w1 = (const float*)d_in[27];
  const float* mlp_b1 = (const float*)d_in[28];
  const float* mlp_w2 = (const float*)d_in[29];
  const float* mlp_b2 = (const float*)d_in[30];
  float* out = (float*)d_out;

  // workspace carve (floats)
  float* ws = (float*)d_ws;
  float* net = ws;  ws += NNODES * DIM;
  float* X0 = ws;   ws += (size_t)BATCH * SEQ * 40;
  float* out1 = ws; ws += (size_t)BATCH * SEQ * 200;
  float* out2 = ws; ws += (size_t)BATCH * SEQ * 200;
  const int Kx[4] = {40, 40, 200, 200};
  float* WihT[4];
  for (int p = 0; p < 4; ++p) { WihT[p] = ws; ws += (size_t)Kx[p] * GATE; }

  hipLaunchKernelGGL(gcn_kernel, dim3(1), dim3(128), 0, stream,
                     emb_w, edge_index, gcn_w1, gcn_b1, gcn_w2, gcn_b2, net);
  const int total = BATCH * SEQ;
  hipLaunchKernelGGL(build_x0, dim3((total + 255) / 256), dim3(256), 0, stream,
                     feats, ids, fc_w, fc_b, net, X0, total);
  for (int p = 0; p < 4; ++p) {
    int n1 = GATE * Kx[p];
    hipLaunchKernelGGL(transpose_kernel, dim3((n1 + 255) / 256), dim3(256), 0, stream,
                       Wih[p], WihT[p], GATE, Kx[p]);
  }
  const size_t smem_bytes = (size_t)SCAN_SMEM_FLOATS * sizeof(float);  // ~200 KB of 320 KB LDS
  // layer 1 (input dim 40): fwd+bwd in one launch (8 blocks: 4 strips x 2 dirs)
  hipLaunchKernelGGL((lstm_scan<40>), dim3(8), dim3(SCAN_THREADS), smem_bytes, stream,
                     X0, WihT[0], Whh[0], bih[0], bhh[0],
                     WihT[1], Whh[1], bih[1], bhh[1], out1);
  // layer 2 (input dim 200)
  hipLaunchKernelGGL((lstm_scan<200>), dim3(8), dim3(SCAN_THREADS), smem_bytes, stream,
                     out1, WihT[2], Whh[2], bih[2], bhh[2],
                     WihT[3], Whh[3], bih[3], bhh[3], out2);
  hipLaunchKernelGGL(mlp_kernel, dim3(total / 16), dim3(128), 0, stream,
                     out2, mlp_w1, mlp_b1, mlp_w2, mlp_b2, out);
}